// Phase2Edges_44538810860115
// MI455X (gfx1250) — compile-verified
//
#include <hip/hip_runtime.h>
#include <stdint.h>

#define NN 8192
#define KTOP 262144u
#define NB1 32768           // level-1 bins: top 15 bits of key (128 KB LDS histogram)
#define NB2 131072          // level-2 bins: low 17 bits of key (global, sparse hits)
#define CAP_EQ 16384u

// workspace layout (uint32 units)
#define OFF_H1 0
#define OFF_H2 (OFF_H1 + NB1)
#define OFF_SC (OFF_H2 + NB2)
#define OFF_EQ (OFF_SC + 16)
#define ZERO_TOT (OFF_SC + 16)
// scalars: sc[0]=B1 sc[1]=countAbove1 sc[2]=keyThresh sc[3]=R(equals kept) sc[4]=eqCount

typedef float v2f __attribute__((ext_vector_type(2)));
typedef float v8f __attribute__((ext_vector_type(8)));

// order-preserving float -> u32 key (larger float => larger key)
__device__ __forceinline__ uint32_t fkey(float v) {
  uint32_t u = __float_as_uint(v);
  return (u & 0x80000000u) ? ~u : (u | 0x80000000u);
}

__global__ void k_zero(uint32_t* __restrict__ ws, int n) {
  int i = blockIdx.x * blockDim.x + threadIdx.x;
  if (i < n) ws[i] = 0u;
}

// Level-1 histogram: 32768 bins in 128 KB dynamic LDS (CDNA5 WGP LDS = 320 KB)
__global__ void k_hist1(const float* __restrict__ p, uint32_t* __restrict__ h1) {
  extern __shared__ uint32_t lh[];
  const int tid = threadIdx.x;
  for (int b = tid; b < NB1; b += 256) lh[b] = 0u;
  __syncthreads();
  const int row0 = blockIdx.x * 64;
  for (int r = 0; r < 64; ++r) {
    const int i = row0 + r;
    const float pi = p[i];
    for (int j = tid; j < NN; j += 256) {
      if (j == i) continue;
      atomicAdd(&lh[fkey(pi * p[j]) >> 17], 1u);
    }
  }
  __syncthreads();
  for (int b = tid; b < NB1; b += 256) {
    uint32_t c = lh[b];
    if (c) atomicAdd(&h1[b], c);
  }
}

__global__ void k_scan1(const uint32_t* __restrict__ h1, uint32_t* __restrict__ sc) {
  __shared__ uint32_t csum[256];
  const int tid = threadIdx.x;
  uint32_t s = 0;
  const int base = tid * 128;
  for (int b = 0; b < 128; ++b) s += h1[base + b];
  csum[tid] = s;
  __syncthreads();
  if (tid == 0) {
    uint32_t acc = 0;
    int c = 255;
    while (c > 0 && acc + csum[c] < KTOP) { acc += csum[c]; --c; }
    int b = c * 128 + 127;
    while (b > c * 128 && acc + h1[b] < KTOP) { acc += h1[b]; --b; }
    sc[0] = (uint32_t)b;  // level-1 threshold bin
    sc[1] = acc;          // elements strictly above bin B1
  }
}

// Level-2 histogram: only elements landing in bin B1 (few), global atomics
__global__ void k_hist2(const float* __restrict__ p, uint32_t* __restrict__ h2,
                        const uint32_t* __restrict__ sc) {
  const uint32_t B1 = sc[0];
  const int tid = threadIdx.x;
  const int row0 = blockIdx.x * 16;
  for (int r = 0; r < 16; ++r) {
    const int i = row0 + r;
    const float pi = p[i];
    for (int j = tid; j < NN; j += 256) {
      if (j == i) continue;
      uint32_t k = fkey(pi * p[j]);
      if ((k >> 17) == B1) atomicAdd(&h2[k & 0x1FFFFu], 1u);
    }
  }
}

__global__ void k_scan2(const uint32_t* __restrict__ h2, uint32_t* __restrict__ sc) {
  __shared__ uint32_t csum[256];
  const int tid = threadIdx.x;
  uint32_t s = 0;
  const int base = tid * 512;
  for (int b = 0; b < 512; ++b) s += h2[base + b];
  csum[tid] = s;
  __syncthreads();
  if (tid == 0) {
    const uint32_t R1 = KTOP - sc[1];
    uint32_t acc = 0;
    int c = 255;
    while (c > 0 && acc + csum[c] < R1) { acc += csum[c]; --c; }
    int b = c * 512 + 511;
    while (b > c * 512 && acc + h2[b] < R1) { acc += h2[b]; --b; }
    sc[2] = (sc[0] << 17) | (uint32_t)b;  // exact 32-bit key threshold
    sc[3] = R1 - acc;                      // how many exact-ties to keep (lowest flat idx first)
  }
}

// Mask write: one wave per 16x16 tile via V_WMMA_F32_16X16X4_F32 (K=0 rank-1 update).
// EXEC all-ones at the WMMA: uniform control flow, exact grid.
__global__ void k_mask(const float* __restrict__ p, float* __restrict__ out,
                       const uint32_t* __restrict__ sc) {
  const uint32_t keyT = sc[2];
  const int lane = threadIdx.x & 31;
  const int wave = threadIdx.x >> 5;
  const int tile = blockIdx.x * 8 + wave;      // 512*512 tiles total
  const int ti = tile >> 9;
  const int tj = tile & 511;
  const int row0 = ti * 16, col0 = tj * 16;
  const int l15 = lane & 15;
  const float pr = p[row0 + l15];
  const float pc = p[col0 + l15];
  const bool lo = (lane < 16);
  // A(16x4): lanes 0-15 hold K=0,1; K=0 column = p rows, rest zero.
  // B(4x16): VGPR0 lanes 0-15 = row K=0 = p cols, rest zero.
  v2f a, b;
  a[0] = lo ? pr : 0.0f; a[1] = 0.0f;
  b[0] = lo ? pc : 0.0f; b[1] = 0.0f;
  v8f c = {};
  v8f d = __builtin_amdgcn_wmma_f32_16x16x4_f32(
      false, a, false, b, (short)0, c, false, false);
  // C/D layout: lanes 0-15 -> M=r, N=lane; lanes 16-31 -> M=8+r, N=lane-16
  const int rbase = row0 + (lo ? 0 : 8);
  const int col = col0 + l15;
#pragma unroll
  for (int r = 0; r < 8; ++r) {
    const int row = rbase + r;
    float o = (fkey(d[r]) > keyT) ? 1.0f : 0.0f;
    if (row == col) o = 0.0f;                  // diagonal excluded
    out[(size_t)row * NN + col] = o;
  }
}

// Collect flat indices of elements exactly equal to the threshold key
__global__ void k_collect(const float* __restrict__ p, uint32_t* __restrict__ sc,
                          uint32_t* __restrict__ eqList) {
  const uint32_t keyT = sc[2];
  const int tid = threadIdx.x;
  const int row0 = blockIdx.x * 16;
  for (int r = 0; r < 16; ++r) {
    const int i = row0 + r;
    const float pi = p[i];
    for (int j = tid; j < NN; j += 256) {
      if (j == i) continue;
      if (fkey(pi * p[j]) == keyT) {
        uint32_t pos = atomicAdd(&sc[4], 1u);
        if (pos < CAP_EQ) eqList[pos] = (uint32_t)i * (uint32_t)NN + (uint32_t)j;
      }
    }
  }
}

// Keep the R ties with smallest flat index (matches top_k tie-break); rank by value
// so the unordered atomic append order does not matter (deterministic).
__global__ void k_resolve(const uint32_t* __restrict__ sc, const uint32_t* __restrict__ eqList,
                          float* __restrict__ out) {
  const uint32_t cnt = sc[4];
  const uint32_t E = (cnt < CAP_EQ) ? cnt : CAP_EQ;
  const uint32_t R = sc[3];
  for (uint32_t e = threadIdx.x; e < E; e += 256) {
    const uint32_t idx = eqList[e];
    uint32_t rank = 0;
    for (uint32_t f = 0; f < E; ++f) rank += (eqList[f] < idx) ? 1u : 0u;
    if (rank < R) out[idx] = 1.0f;
  }
}

extern "C" void kernel_launch(void* const* d_in, const int* in_sizes, int n_in,
                              void* d_out, int out_size, void* d_ws, size_t ws_size,
                              hipStream_t stream) {
  (void)in_sizes; (void)n_in; (void)out_size; (void)ws_size;
  const float* p = (const float*)d_in[0];
  float* out = (float*)d_out;
  uint32_t* ws = (uint32_t*)d_ws;
  uint32_t* h1 = ws + OFF_H1;
  uint32_t* h2 = ws + OFF_H2;
  uint32_t* sc = ws + OFF_SC;
  uint32_t* eq = ws + OFF_EQ;

  k_zero<<<(ZERO_TOT + 255) / 256, 256, 0, stream>>>(ws, ZERO_TOT);
  k_hist1<<<NN / 64, 256, NB1 * sizeof(uint32_t), stream>>>(p, h1);   // 128 KB LDS/WG
  k_scan1<<<1, 256, 0, stream>>>(h1, sc);
  k_hist2<<<NN / 16, 256, 0, stream>>>(p, h2, sc);
  k_scan2<<<1, 256, 0, stream>>>(h2, sc);
  k_mask<<<(512 * 512) / 8, 256, 0, stream>>>(p, out, sc);            // WMMA pass
  k_collect<<<NN / 16, 256, 0, stream>>>(p, sc, eq);
  k_resolve<<<1, 256, 0, stream>>>(sc, eq, out);
}